// GraspCVAELoss_20512763806172
// MI455X (gfx1250) — compile-verified
//
#include <hip/hip_runtime.h>
#include <math.h>

// ---- problem constants (from reference) ----
#define BATCH 32
#define P1    778      // hand verts
#define NF    1538     // faces
#define P2    3000     // obj points
#define PD    61
#define LAT   64
#define MTILES 49      // ceil(778/16)
#define NTILES 188     // ceil(3000/16)
#define BIGF  3.0e38f

typedef float v2f __attribute__((ext_vector_type(2)));
typedef float v8f __attribute__((ext_vector_type(8)));

// monotone float -> uint transform: a < b  <=>  ord(a) < ord(b) (unsigned)
__device__ inline unsigned float_ordered(float f) {
  unsigned u = __float_as_uint(f);
  return u ^ ((unsigned)((int)u >> 31) | 0x80000000u);
}

// ---------------------------------------------------------------------------
// block reduction (blockDim.x == 256)
// ---------------------------------------------------------------------------
__device__ inline float block_reduce_256(float v) {
  __shared__ float sh[256];
  __syncthreads();
  sh[threadIdx.x] = v;
  __syncthreads();
  for (int s = 128; s > 0; s >>= 1) {
    if ((int)threadIdx.x < s) sh[threadIdx.x] += sh[threadIdx.x + s];
    __syncthreads();
  }
  return sh[0];
}

// ---------------------------------------------------------------------------
// 1) trivial elementwise sums: param, recon, KLD
// ---------------------------------------------------------------------------
__global__ __launch_bounds__(256)
void simple_sums(const float* __restrict__ rx, const float* __restrict__ x,
                 const float* __restrict__ mu, const float* __restrict__ lv,
                 const float* __restrict__ rxyz, const float* __restrict__ hxyz,
                 float* __restrict__ sums) {
  int tid = blockIdx.x * blockDim.x + threadIdx.x;
  int stride = gridDim.x * blockDim.x;
  float p = 0.f, r = 0.f, k = 0.f;
  for (int i = tid; i < BATCH * PD; i += stride) {
    float d = rx[i] - x[i];
    p += d * d;
  }
  for (int i = tid; i < BATCH * P1 * 3; i += stride) {
    float d = rxyz[i] - hxyz[i];
    r += d * d;
  }
  for (int i = tid; i < BATCH * LAT; i += stride) {
    float l = lv[i], m = mu[i];
    k += 1.0f + l - m * m - expf(l);
  }
  p = block_reduce_256(p);
  r = block_reduce_256(r);
  k = block_reduce_256(k);
  if (threadIdx.x == 0) {
    atomicAdd(&sums[0], p);
    atomicAdd(&sums[1], r);
    atomicAdd(&sums[2], k);
  }
}

// ---------------------------------------------------------------------------
// 2) face-normal scatter for both vertex sets (normals buffers pre-zeroed)
// ---------------------------------------------------------------------------
__device__ inline void atomic_add3(float* p, float a, float b, float c) {
  atomicAdd(p + 0, a);
  atomicAdd(p + 1, b);
  atomicAdd(p + 2, c);
}

__global__ __launch_bounds__(256)
void face_normals(const float* __restrict__ vr, const float* __restrict__ vg,
                  const int* __restrict__ faces,
                  float* __restrict__ nr, float* __restrict__ ng) {
  int i = blockIdx.x * blockDim.x + threadIdx.x;
  if (i >= BATCH * NF) return;
  int b = i / NF;
  const int* fp = faces + (size_t)i * 3;
  int i0 = fp[0], i1 = fp[1], i2 = fp[2];
  size_t base = (size_t)b * P1 * 3;

  {
    const float* V = vr + base;
    float ax = V[i1*3+0]-V[i0*3+0], ay = V[i1*3+1]-V[i0*3+1], az = V[i1*3+2]-V[i0*3+2];
    float bx = V[i2*3+0]-V[i0*3+0], by = V[i2*3+1]-V[i0*3+1], bz = V[i2*3+2]-V[i0*3+2];
    float cx = ay*bz - az*by, cy = az*bx - ax*bz, cz = ax*by - ay*bx;
    float* N = nr + base;
    atomic_add3(N + i0*3, cx, cy, cz);
    atomic_add3(N + i1*3, cx, cy, cz);
    atomic_add3(N + i2*3, cx, cy, cz);
  }
  {
    const float* V = vg + base;
    float ax = V[i1*3+0]-V[i0*3+0], ay = V[i1*3+1]-V[i0*3+1], az = V[i1*3+2]-V[i0*3+2];
    float bx = V[i2*3+0]-V[i0*3+0], by = V[i2*3+1]-V[i0*3+1], bz = V[i2*3+2]-V[i0*3+2];
    float cx = ay*bz - az*by, cy = az*bx - ax*bz, cz = ax*by - ay*bx;
    float* N = ng + base;
    atomic_add3(N + i0*3, cx, cy, cz);
    atomic_add3(N + i1*3, cx, cy, cz);
    atomic_add3(N + i2*3, cx, cy, cz);
  }
}

__global__ __launch_bounds__(256)
void normalize_normals(float* __restrict__ nr, float* __restrict__ ng) {
  int i = blockIdx.x * blockDim.x + threadIdx.x;
  if (i >= BATCH * P1) return;
  float* bufs[2] = {nr, ng};
  for (int t = 0; t < 2; ++t) {
    float* p = bufs[t] + (size_t)i * 3;
    float x = p[0], y = p[1], z = p[2];
    float n = sqrtf(x * x + y * y + z * z);
    float s = 1.0f / fmaxf(n, 1e-6f);
    p[0] = x * s; p[1] = y * s; p[2] = z * s;
  }
}

// ---------------------------------------------------------------------------
// final cross-lane argmin over a 16-lane half (u64 keys, branchless)
// ---------------------------------------------------------------------------
__device__ inline unsigned long long halfwave_min_u64(unsigned long long k) {
#pragma unroll
  for (int off = 8; off > 0; off >>= 1) {
    unsigned lo = (unsigned)k, hi = (unsigned)(k >> 32);
    unsigned olo = __shfl_xor(lo, off, 32);
    unsigned ohi = __shfl_xor(hi, off, 32);
    unsigned long long ok = ((unsigned long long)ohi << 32) | olo;
    k = ok < k ? ok : k;
  }
  return k;
}

// ---------------------------------------------------------------------------
// 3) WMMA row pass: for each hand vert, nearest obj point -> h2o = |xp - near|
//    One wave per (batch, 16-row tile). D tile layout: VGPR j -> row j+8*half,
//    lane&15 -> column. Per-lane running argmin key, reduced once at the end.
//    B-tile loads are software-pipelined (prefetch next tile before compute).
// ---------------------------------------------------------------------------
__global__ __launch_bounds__(32)
void nn_rows_wmma(const float* __restrict__ hand, const float* __restrict__ obj,
                  float* __restrict__ h2o) {
  const int b    = blockIdx.y;
  const int mt   = blockIdx.x;
  const int lane = threadIdx.x;
  const int half = lane >> 4;
  const int l16  = lane & 15;

  const float* Hb = hand + (size_t)b * (P1 * 3);
  const float* Ob = obj  + (size_t)b * (P2 * 3);

  // A-matrix 16x4: lanes 0-15 hold (K0,K1)=(x,y); lanes 16-31 hold (K2,K3)=(z,0)
  int m  = mt * 16 + l16;
  int mc = m < P1 ? m : 0;
  float hx = Hb[mc*3+0], hy = Hb[mc*3+1], hz = Hb[mc*3+2];
  v2f a;
  a.x = half ? hz : hx;
  a.y = half ? 0.f : hy;

  unsigned long long bestk[8];
#pragma unroll
  for (int j = 0; j < 8; ++j) bestk[j] = ~0ull;

  // prefetch tile 0
  float ox = Ob[l16*3+0], oy = Ob[l16*3+1], oz = Ob[l16*3+2];

#pragma unroll 2
  for (int nt = 0; nt < NTILES; ++nt) {
    int n = nt * 16 + l16;
    float cx = ox, cy = oy, cz = oz;
    // prefetch next tile (clamped) while current tile computes
    int nn  = n + 16;
    int nncl = nn < P2 ? nn : 0;
    ox = Ob[nncl*3+0]; oy = Ob[nncl*3+1]; oz = Ob[nncl*3+2];

    float y2p = cx*cx + cy*cy + cz*cz + ((n < P2) ? 0.f : BIGF);
    // B-matrix 4x16: lanes 0-15 hold (K0,K1); lanes 16-31 hold (K2,K3)
    v2f bb;
    bb.x = half ? cz : cx;
    bb.y = half ? 0.f : cy;

    v8f c = {};
    c = __builtin_amdgcn_wmma_f32_16x16x4_f32(false, a, false, bb, (short)0, c,
                                              false, false);
#pragma unroll
    for (int j = 0; j < 8; ++j) {
      float val = fmaf(-2.0f, c[j], y2p);
      unsigned long long key =
          ((unsigned long long)float_ordered(val) << 32) | (unsigned)n;
      bestk[j] = key < bestk[j] ? key : bestk[j];
    }
  }

#pragma unroll
  for (int j = 0; j < 8; ++j) {
    unsigned long long k = halfwave_min_u64(bestk[j]);
    if (l16 == j) {
      int mr = mt * 16 + j + half * 8;
      if (mr < P1) {
        int ni = (int)(unsigned)k;
        float dx = Hb[mr*3+0] - Ob[ni*3+0];
        float dy = Hb[mr*3+1] - Ob[ni*3+1];
        float dz = Hb[mr*3+2] - Ob[ni*3+2];
        h2o[(size_t)b * P1 + mr] = sqrtf(dx*dx + dy*dy + dz*dz);
      }
    }
  }
}

// ---------------------------------------------------------------------------
// 4) WMMA column pass: for each obj point, nearest hand vert -> signed o2h
// ---------------------------------------------------------------------------
__global__ __launch_bounds__(32)
void nn_cols_wmma(const float* __restrict__ hand, const float* __restrict__ obj,
                  const float* __restrict__ normals, float* __restrict__ o2h) {
  const int b    = blockIdx.y;
  const int nt   = blockIdx.x;
  const int lane = threadIdx.x;
  const int half = lane >> 4;
  const int l16  = lane & 15;

  const float* Hb = hand    + (size_t)b * (P1 * 3);
  const float* Ob = obj     + (size_t)b * (P2 * 3);
  const float* Nb = normals + (size_t)b * (P1 * 3);

  int q  = nt * 16 + l16;
  int qc = q < P2 ? q : 0;
  float qx = Ob[qc*3+0], qy = Ob[qc*3+1], qz = Ob[qc*3+2];
  v2f a;
  a.x = half ? qz : qx;
  a.y = half ? 0.f : qy;

  unsigned long long bestk[8];
#pragma unroll
  for (int j = 0; j < 8; ++j) bestk[j] = ~0ull;

  // prefetch tile 0
  float hx = Hb[l16*3+0], hy = Hb[l16*3+1], hz = Hb[l16*3+2];

#pragma unroll 2
  for (int mt = 0; mt < MTILES; ++mt) {
    int p = mt * 16 + l16;
    float cx = hx, cy = hy, cz = hz;
    // prefetch next tile (clamped) while current tile computes
    int np  = p + 16;
    int npcl = np < P1 ? np : 0;
    hx = Hb[npcl*3+0]; hy = Hb[npcl*3+1]; hz = Hb[npcl*3+2];

    float x2p = cx*cx + cy*cy + cz*cz + ((p < P1) ? 0.f : BIGF);
    v2f bb;
    bb.x = half ? cz : cx;
    bb.y = half ? 0.f : cy;

    v8f c = {};
    c = __builtin_amdgcn_wmma_f32_16x16x4_f32(false, a, false, bb, (short)0, c,
                                              false, false);
#pragma unroll
    for (int j = 0; j < 8; ++j) {
      float val = fmaf(-2.0f, c[j], x2p);
      unsigned long long key =
          ((unsigned long long)float_ordered(val) << 32) | (unsigned)p;
      bestk[j] = key < bestk[j] ? key : bestk[j];
    }
  }

#pragma unroll
  for (int j = 0; j < 8; ++j) {
    unsigned long long k = halfwave_min_u64(bestk[j]);
    if (l16 == j) {
      int qr = nt * 16 + j + half * 8;
      if (qr < P2) {
        int pi = (int)(unsigned)k;
        float yx = Ob[qr*3+0] - Hb[pi*3+0];
        float yy = Ob[qr*3+1] - Hb[pi*3+1];
        float yz = Ob[qr*3+2] - Hb[pi*3+2];
        float d  = Nb[pi*3+0]*yx + Nb[pi*3+1]*yy + Nb[pi*3+2]*yz;
        float sgn = (d > 0.f) ? 1.f : ((d < 0.f) ? -1.f : 0.f);
        o2h[(size_t)b * P2 + qr] = sqrtf(yx*yx + yy*yy + yz*yz) * sgn;
      }
    }
  }
}

// ---------------------------------------------------------------------------
// 5) weighted loss terms
// ---------------------------------------------------------------------------
__global__ __launch_bounds__(256)
void loss_terms(const float* __restrict__ h2o, const float* __restrict__ h2o_g,
                const float* __restrict__ o2h, const float* __restrict__ o2h_g,
                const float* __restrict__ vw, float* __restrict__ sums) {
  int tid = blockIdx.x * blockDim.x + threadIdx.x;
  int stride = gridDim.x * blockDim.x;
  float sh = 0.f, so = 0.f;
  for (int i = tid; i < BATCH * P1; i += stride) {
    int mm = i % P1;
    float w2 = powf(vw[mm], 0.4f);  // 1/2.5
    sh += fabsf(fabsf(h2o[i]) - fabsf(h2o_g[i])) * w2;
  }
  for (int i = tid; i < BATCH * P2; i += stride) {
    float o = o2h[i], og = o2h_g[i];
    float w = (o < 0.f) ? 1.5f : (((og < 0.01f) && (og > -0.005f)) ? 1.0f : 0.1f);
    so += fabsf(o - og) * w;
  }
  sh = block_reduce_256(sh);
  so = block_reduce_256(so);
  if (threadIdx.x == 0) {
    atomicAdd(&sums[3], sh);
    atomicAdd(&sums[4], so);
  }
}

// ---------------------------------------------------------------------------
// 6) finalize the 5 scalar outputs
// ---------------------------------------------------------------------------
__global__ void finalize(const float* __restrict__ sums, float* __restrict__ out) {
  if (threadIdx.x == 0 && blockIdx.x == 0) {
    float param = sums[0] / (float)BATCH;
    float recon = sums[1] / (float)BATCH;
    float kld   = -0.5f * sums[2] / (float)BATCH;
    float lh = 35.0f * 0.995f * sums[3] / (float)(BATCH * P1);
    float lo = 30.0f * 0.995f * sums[4] / (float)(BATCH * P2);
    float ho = lh + lo;
    float loss = (recon + kld) + 0.1f * param + 10.0f * ho;
    out[0] = loss;
    out[1] = param;
    out[2] = ho;
    out[3] = recon;
    out[4] = kld;
  }
}

// ---------------------------------------------------------------------------
extern "C" void kernel_launch(void* const* d_in, const int* in_sizes, int n_in,
                              void* d_out, int out_size, void* d_ws, size_t ws_size,
                              hipStream_t stream) {
  (void)in_sizes; (void)n_in; (void)out_size; (void)ws_size;

  const float* recon_x = (const float*)d_in[0];
  const float* x       = (const float*)d_in[1];
  const float* mu      = (const float*)d_in[2];
  const float* logvar  = (const float*)d_in[3];
  const float* rxyz    = (const float*)d_in[4];
  const float* hxyz    = (const float*)d_in[5];
  const int*   faces   = (const int*)d_in[6];
  const float* obj     = (const float*)d_in[7];
  const float* vw      = (const float*)d_in[8];
  float* out = (float*)d_out;

  float* wsf   = (float*)d_ws;
  float* sums  = wsf;                       // [0]param [1]recon [2]kld [3]lh [4]lo
  float* h2o   = wsf + 8;                   // BATCH*P1
  float* h2o_g = h2o + BATCH * P1;
  float* o2h   = h2o_g + BATCH * P1;        // BATCH*P2
  float* o2h_g = o2h + BATCH * P2;
  float* nrm   = o2h_g + BATCH * P2;        // BATCH*P1*3
  float* nrm_g = nrm + BATCH * P1 * 3;
  size_t used = (size_t)(8 + 2 * BATCH * P1 + 2 * BATCH * P2 + 2 * BATCH * P1 * 3)
                * sizeof(float);

  hipMemsetAsync(d_ws, 0, used, stream);

  simple_sums<<<(BATCH * P1 * 3 + 255) / 256, 256, 0, stream>>>(
      recon_x, x, mu, logvar, rxyz, hxyz, sums);

  face_normals<<<(BATCH * NF + 255) / 256, 256, 0, stream>>>(
      rxyz, hxyz, faces, nrm, nrm_g);
  normalize_normals<<<(BATCH * P1 + 255) / 256, 256, 0, stream>>>(nrm, nrm_g);

  dim3 gR(MTILES, BATCH);
  nn_rows_wmma<<<gR, 32, 0, stream>>>(rxyz, obj, h2o);
  nn_rows_wmma<<<gR, 32, 0, stream>>>(hxyz, obj, h2o_g);

  dim3 gC(NTILES, BATCH);
  nn_cols_wmma<<<gC, 32, 0, stream>>>(rxyz, obj, nrm, o2h);
  nn_cols_wmma<<<gC, 32, 0, stream>>>(hxyz, obj, nrm_g, o2h_g);

  loss_terms<<<(BATCH * P2 + 255) / 256, 256, 0, stream>>>(
      h2o, h2o_g, o2h, o2h_g, vw, sums);
  finalize<<<1, 32, 0, stream>>>(sums, out);
}